// PhaseModulatedMambaBlock_62208306316038
// MI455X (gfx1250) — compile-verified
//
#include <hip/hip_runtime.h>
#include <hip/hip_bf16.h>
#include <math.h>

// ---------------------------------------------------------------------------
// PhaseModulatedMambaBlock on gfx1250 (MI455X)
// GEMMs via V_WMMA_F32_16X16X4_F32 with 32x32 (2x2) per-wave tiles for
// operand reuse; selective scan via one wave per (b, d_inner) row.
// ---------------------------------------------------------------------------

typedef __attribute__((ext_vector_type(2))) float v2f;
typedef __attribute__((ext_vector_type(8))) float v8f;

#define B_SZ    8
#define LSEQ    200
#define DMODEL  256
#define DSTATE  64
#define DCONV   4
#define NBANDS  4
#define DINNER  512
#define DTRANK  16
#define BL      (B_SZ * LSEQ)          // 1600 rows
#define XPCOLS  (DTRANK + 2 * DSTATE)  // 144
#define LN_EPS  1e-12f

__device__ __forceinline__ float softplusf(float x) {
  return (x > 20.f) ? x : log1pf(expf(x));
}
__device__ __forceinline__ float siluf(float x) {
  return x / (1.f + expf(-x));
}

// ---------------------------------------------------------------------------
// Generic WMMA fp32 GEMM, wave tile = (MT*16) x (NT*16).
// A layout (16x4 f32): lanes 0-15 hold rows M=0..15 with K={0,1},
// lanes 16-31 same rows with K={2,3}.  B (4x16) mirrors it.
// EPI: 0 = plain store, 1 = dt epilogue softplus(acc+bias[n])*scale[m],
//      2 = acc + residual.
// ---------------------------------------------------------------------------
template <int MT, int NT, int EPI>
__global__ __launch_bounds__(256) void k_gemm(
    const float* __restrict__ A, const float* __restrict__ B,
    const float* __restrict__ e0, const float* __restrict__ e1,
    float* __restrict__ C, int M, int N, int K, int lda) {
  constexpr int TM = MT * 16, TN = NT * 16;
  int wave = (blockIdx.x * blockDim.x + threadIdx.x) >> 5;
  int tilesN = N / TN;
  int tiles = (M / TM) * tilesN;
  if (wave >= tiles) return;
  int lane = threadIdx.x & 31;
  int l16 = lane & 15, half = lane >> 4;
  int m0 = (wave / tilesN) * TM;
  int n0 = (wave % tilesN) * TN;

  v8f c[MT][NT];
#pragma unroll
  for (int mi = 0; mi < MT; ++mi)
#pragma unroll
    for (int ni = 0; ni < NT; ++ni)
      c[mi][ni] = (v8f){0.f, 0.f, 0.f, 0.f, 0.f, 0.f, 0.f, 0.f};

  const float* __restrict__ a0[MT];
  const float* __restrict__ b0[NT];
#pragma unroll
  for (int mi = 0; mi < MT; ++mi) {
    a0[mi] = A + (size_t)(m0 + mi * 16 + l16) * lda + 2 * half;
    __builtin_prefetch(a0[mi], 0, 0);  // global_prefetch_b8: seed strided A stream
  }
#pragma unroll
  for (int ni = 0; ni < NT; ++ni)
    b0[ni] = B + (size_t)(2 * half) * N + n0 + ni * 16 + l16;

  for (int k = 0; k < K; k += 4) {
    v2f av[MT], bv[NT];
#pragma unroll
    for (int mi = 0; mi < MT; ++mi) {
      av[mi][0] = a0[mi][k];
      av[mi][1] = a0[mi][k + 1];
    }
#pragma unroll
    for (int ni = 0; ni < NT; ++ni) {
      bv[ni][0] = b0[ni][(size_t)k * N];
      bv[ni][1] = b0[ni][(size_t)(k + 1) * N];
    }
#pragma unroll
    for (int mi = 0; mi < MT; ++mi)
#pragma unroll
      for (int ni = 0; ni < NT; ++ni)
        c[mi][ni] = __builtin_amdgcn_wmma_f32_16x16x4_f32(
            false, av[mi], false, bv[ni], (short)0, c[mi][ni], false, false);
  }

#pragma unroll
  for (int mi = 0; mi < MT; ++mi)
#pragma unroll
    for (int ni = 0; ni < NT; ++ni) {
      int col = n0 + ni * 16 + l16;
#pragma unroll
      for (int i = 0; i < 8; ++i) {
        int row = m0 + mi * 16 + i + 8 * half;
        size_t idx = (size_t)row * N + col;
        float v = c[mi][ni][i];
        if (EPI == 1) v = softplusf(v + e0[col]) * e1[row];
        if (EPI == 2) v = v + e0[idx];
        C[idx] = v;
      }
    }
}

// ---------------------------------------------------------------------------
// Depthwise causal conv (width 4) + SiLU.  xz layout: (BL, 1024); x_m = [:,:512]
// ---------------------------------------------------------------------------
__global__ __launch_bounds__(256) void k_conv_silu(
    const float* __restrict__ xz, const float* __restrict__ conv_w,
    const float* __restrict__ conv_b, float* __restrict__ xm) {
  int idx = blockIdx.x * blockDim.x + threadIdx.x;
  if (idx >= BL * DINNER) return;
  int d = idx % DINNER;
  int row = idx / DINNER;           // b*L + l
  int l = row % LSEQ;
  int base = row - l;               // b*L
  float acc = conv_b[d];
#pragma unroll
  for (int j = 0; j < DCONV; ++j) {
    int lj = l - (DCONV - 1) + j;
    if (lj >= 0)
      acc += conv_w[d * DCONV + j] * xz[(size_t)(base + lj) * (2 * DINNER) + d];
  }
  xm[(size_t)row * DINNER + d] = siluf(acc);
}

// scale[b*L+l] = 1 + softplus(mean over bands of delta_mag[b, band, l])
__global__ __launch_bounds__(256) void k_scale(
    const float* __restrict__ delta_mag, float* __restrict__ scale) {
  int idx = blockIdx.x * blockDim.x + threadIdx.x;
  if (idx >= BL) return;
  int b = idx / LSEQ, l = idx % LSEQ;
  float s = 0.f;
#pragma unroll
  for (int j = 0; j < NBANDS; ++j)
    s += delta_mag[(size_t)b * NBANDS * LSEQ + j * LSEQ + l];
  s *= (1.f / NBANDS);
  scale[idx] = 1.f + softplusf(s);
}

// C_mod[row, n] = xp[row, 80+n] + phase_feats[row, :] @ W_phase[:, n]
__global__ __launch_bounds__(256) void k_cmod(
    const float* __restrict__ xp, const float* __restrict__ cos_phi,
    const float* __restrict__ sin_phi, const float* __restrict__ W_phase,
    float* __restrict__ cmod) {
  int idx = blockIdx.x * blockDim.x + threadIdx.x;
  if (idx >= BL * DSTATE) return;
  int n = idx % DSTATE;
  int row = idx / DSTATE;
  int b = row / LSEQ, l = row % LSEQ;
  float acc = xp[(size_t)row * XPCOLS + DTRANK + DSTATE + n];
#pragma unroll
  for (int k = 0; k < NBANDS; ++k) {
    float cp = cos_phi[(size_t)b * NBANDS * LSEQ + k * LSEQ + l];
    float sph = sin_phi[(size_t)b * NBANDS * LSEQ + k * LSEQ + l];
    acc += cp * W_phase[k * DSTATE + n];
    acc += sph * W_phase[(NBANDS + k) * DSTATE + n];
  }
  cmod[(size_t)row * DSTATE + n] = acc;
}

// ---------------------------------------------------------------------------
// Selective scan: one wave per (b, d). Lane holds states n and n+32.
// y_gated[row,d] = (sum_n h*C + u*D_skip[d]) * silu(z)
// ---------------------------------------------------------------------------
__global__ __launch_bounds__(256) void k_scan(
    const float* __restrict__ dtbuf, const float* __restrict__ xm,
    const float* __restrict__ xz, const float* __restrict__ xp,
    const float* __restrict__ cmod, const float* __restrict__ A_log,
    const float* __restrict__ D_skip, float* __restrict__ yg) {
  int wave = (blockIdx.x * blockDim.x + threadIdx.x) >> 5;
  if (wave >= B_SZ * DINNER) return;
  int lane = threadIdx.x & 31;
  int b = wave / DINNER;
  int d = wave % DINNER;
  int n0 = lane, n1 = lane + 32;

  float A0 = -expf(A_log[(size_t)d * DSTATE + n0]);
  float A1 = -expf(A_log[(size_t)d * DSTATE + n1]);
  float Dd = D_skip[d];
  float h0 = 0.f, h1 = 0.f;

  for (int t = 0; t < LSEQ; ++t) {
    int row = b * LSEQ + t;
    float dt = dtbuf[(size_t)row * DINNER + d];
    float u  = xm[(size_t)row * DINNER + d];
    float z  = xz[(size_t)row * (2 * DINNER) + DINNER + d];
    float bs0 = xp[(size_t)row * XPCOLS + DTRANK + n0];
    float bs1 = xp[(size_t)row * XPCOLS + DTRANK + n1];
    float c0 = cmod[(size_t)row * DSTATE + n0];
    float c1 = cmod[(size_t)row * DSTATE + n1];

    h0 = expf(dt * A0) * h0 + dt * bs0 * u;
    h1 = expf(dt * A1) * h1 + dt * bs1 * u;
    float p = h0 * c0 + h1 * c1;
#pragma unroll
    for (int off = 16; off > 0; off >>= 1)
      p += __shfl_down(p, off, 32);
    if (lane == 0)
      yg[(size_t)row * DINNER + d] = (p + u * Dd) * siluf(z);
  }
}

// ---------------------------------------------------------------------------
// LayerNorm over D_MODEL per row; one 256-thread block per row.
// ---------------------------------------------------------------------------
__global__ __launch_bounds__(256) void k_layernorm(
    const float* __restrict__ pre, const float* __restrict__ gamma,
    const float* __restrict__ beta, float* __restrict__ out) {
  __shared__ float red[256];
  int row = blockIdx.x;
  int tid = threadIdx.x;
  float v = pre[(size_t)row * DMODEL + tid];

  red[tid] = v;
  __syncthreads();
  for (int s = 128; s > 0; s >>= 1) {
    if (tid < s) red[tid] += red[tid + s];
    __syncthreads();
  }
  float mu = red[0] * (1.f / DMODEL);
  __syncthreads();

  float dv = v - mu;
  red[tid] = dv * dv;
  __syncthreads();
  for (int s = 128; s > 0; s >>= 1) {
    if (tid < s) red[tid] += red[tid + s];
    __syncthreads();
  }
  float var = red[0] * (1.f / DMODEL);
  out[(size_t)row * DMODEL + tid] = dv * rsqrtf(var + LN_EPS) * gamma[tid] + beta[tid];
}

// ---------------------------------------------------------------------------
extern "C" void kernel_launch(void* const* d_in, const int* in_sizes, int n_in,
                              void* d_out, int out_size, void* d_ws, size_t ws_size,
                              hipStream_t stream) {
  const float* x         = (const float*)d_in[0];
  const float* cos_phi   = (const float*)d_in[1];
  const float* sin_phi   = (const float*)d_in[2];
  const float* delta_mag = (const float*)d_in[3];
  const float* W_in      = (const float*)d_in[4];
  const float* conv_w    = (const float*)d_in[5];
  const float* conv_b    = (const float*)d_in[6];
  const float* W_x       = (const float*)d_in[7];
  const float* W_dt      = (const float*)d_in[8];
  const float* b_dt      = (const float*)d_in[9];
  const float* A_log     = (const float*)d_in[10];
  const float* D_skip    = (const float*)d_in[11];
  const float* W_out     = (const float*)d_in[12];
  const float* gamma     = (const float*)d_in[13];
  const float* beta      = (const float*)d_in[14];
  const float* W_phase   = (const float*)d_in[15];
  float* out = (float*)d_out;

  // Workspace layout (floats)
  float* ws    = (float*)d_ws;
  float* xz    = ws;                                // 1600*1024
  float* xm    = xz   + (size_t)BL * 2 * DINNER;    // 1600*512
  float* xp    = xm   + (size_t)BL * DINNER;        // 1600*144
  float* dtb   = xp   + (size_t)BL * XPCOLS;        // 1600*512
  float* scale = dtb  + (size_t)BL * DINNER;        // 1600
  float* cmod  = scale + BL;                        // 1600*64
  float* yg    = cmod + (size_t)BL * DSTATE;        // 1600*512
  float* pre   = yg   + (size_t)BL * DINNER;        // 1600*256

  // 1) xz = x @ W_in     (1600 x 256 x 1024), 32x32 wave tiles
  {
    int tiles = (BL / 32) * ((2 * DINNER) / 32);    // 1600 waves
    k_gemm<2, 2, 0><<<(tiles + 7) / 8, 256, 0, stream>>>(
        x, W_in, nullptr, nullptr, xz, BL, 2 * DINNER, DMODEL, DMODEL);
  }
  // 2) depthwise conv + SiLU -> x_m
  k_conv_silu<<<(BL * DINNER + 255) / 256, 256, 0, stream>>>(xz, conv_w, conv_b, xm);
  // 3) xp = x_m @ W_x    (1600 x 512 x 144), 16x48 wave tiles (144 = 3*48)
  {
    int tiles = (BL / 16) * (XPCOLS / 48);          // 300 waves
    k_gemm<1, 3, 0><<<(tiles + 7) / 8, 256, 0, stream>>>(
        xm, W_x, nullptr, nullptr, xp, BL, XPCOLS, DINNER, DINNER);
  }
  // 4) per-row dt scale from delta_mag
  k_scale<<<(BL + 255) / 256, 256, 0, stream>>>(delta_mag, scale);
  // 5) dt = softplus(dt_lr @ W_dt + b_dt) * scale   (A = xp[:, :16], lda=144)
  {
    int tiles = (BL / 32) * (DINNER / 32);          // 800 waves
    k_gemm<2, 2, 1><<<(tiles + 7) / 8, 256, 0, stream>>>(
        xp, W_dt, b_dt, scale, dtb, BL, DINNER, DTRANK, XPCOLS);
  }
  // 6) C_mod = C_st + phase_feats @ W_phase
  k_cmod<<<(BL * DSTATE + 255) / 256, 256, 0, stream>>>(
      xp, cos_phi, sin_phi, W_phase, cmod);
  // 7) selective scan + skip + gate  -> yg
  {
    int waves = B_SZ * DINNER;                      // 4096 waves
    k_scan<<<(waves + 7) / 8, 256, 0, stream>>>(
        dtb, xm, xz, xp, cmod, A_log, D_skip, yg);
  }
  // 8) pre = yg @ W_out + residual  (1600 x 512 x 256)
  {
    int tiles = (BL / 32) * (DMODEL / 32);          // 400 waves
    k_gemm<2, 2, 2><<<(tiles + 7) / 8, 256, 0, stream>>>(
        yg, W_out, x, nullptr, pre, BL, DMODEL, DINNER, DINNER);
  }
  // 9) layernorm -> out
  k_layernorm<<<BL, 256, 0, stream>>>(pre, gamma, beta, out);
}